// _ScaleFFTFilter_45380624449589
// MI455X (gfx1250) — compile-verified
//
#include <hip/hip_runtime.h>

// CDNA5 / gfx1250 wave32 WMMA implementation of
//   y = irfft(rfft(x, ortho) * H, ortho)  ==  Re{ IDFT( DFT(x) * Hfull ) } / T
// via four-step FFT, T = 8192 = 64 * 128, all stages as f32 WMMA matmuls.
// - Hermitian plane symmetry: only k2 in [0,64] planes computed/stored.
// - LDS cos/sin tables with linear phase recurrences (no sincos in loops).
// - N-dimension register blocking: one wave reuses its DFT-row (A) operands
//   across 4 (K1/K3) or 2 (K2) f-column tiles, amortizing all table/index VALU.

typedef float v2f __attribute__((ext_vector_type(2)));
typedef float v8f __attribute__((ext_vector_type(8)));

#define TWO_PI 6.283185307179586f

static constexpr int B_   = 8;
static constexpr int T_   = 8192;
static constexpr int F_   = 512;
static constexpr int FREQ = 4097;
static constexpr int KP   = 65;       // stored k2 planes: 0..64

static __device__ __forceinline__ v8f wmma4(v2f a, v2f b, v8f c) {
  // D = A(16x4 f32) x B(4x16 f32) + C(16x16 f32)
  return __builtin_amdgcn_wmma_f32_16x16x4_f32(false, a, false, b, (short)0, c,
                                               false, false);
}

// ---------------------------------------------------------------------------
// K1: per (b, n1):  Bm[b,n1,k2,f] = w8192^(-n1 k2) * sum_n2 w128^(-n2 k2) x[b, n1+64*n2, f]
//     M = k2 (5 tiles cover 0..79, store 0..64), K = n2 (128), N = 64 f / block
// ---------------------------------------------------------------------------
__global__ __launch_bounds__(160) void k1_fwd128(const float* __restrict__ x,
                                                 float2* __restrict__ buf1) {
  __shared__ float2 w128n[128];         // {cos, -sin}(2*pi*p/128)  (w^-)
  if (threadIdx.x < 128) {
    float s, c;
    __sincosf(TWO_PI * (float)threadIdx.x * (1.0f / 128.0f), &s, &c);
    float2 e; e.x = c; e.y = -s;
    w128n[threadIdx.x] = e;
  }
  __syncthreads();

  const int bn   = blockIdx.x;          // b*64 + n1
  const int b    = bn >> 6;
  const int n1   = bn & 63;
  const int f0   = blockIdx.y * 64;
  const int wave = threadIdx.x >> 5;    // 0..4
  const int lane = threadIdx.x & 31;
  const int half = lane >> 4;
  const int l16  = lane & 15;
  const int m0   = wave * 16;           // k2 tile base
  const int rowA = m0 + l16;            // k2 row for A operand

  const int stepK = (rowA * 4) & 127;   // phase step per K-chunk
  int pa = (rowA * 2 * half) & 127;     // phase of K = 2*half
  int pb = (pa + rowA) & 127;           // phase of K = 2*half + 1

  v8f accR[4] = {};
  v8f accI[4] = {};
#pragma unroll 4
  for (int kk = 0; kk < 32; ++kk) {
    const int n2a = kk * 4 + 2 * half;
    const int n2b = n2a + 1;
    const float2 ea = w128n[pa];
    const float2 eb = w128n[pb];
    pa = (pa + stepK) & 127;
    pb = (pb + stepK) & 127;
    v2f ar; ar.x = ea.x; ar.y = eb.x;   // Re w^-
    v2f ai; ai.x = ea.y; ai.y = eb.y;   // Im w^-  (pre-negated in table)
    const size_t rowOffA = (size_t)(b * T_ + (n2a * 64 + n1)) * F_;
    const size_t rowOffB = (size_t)(b * T_ + (n2b * 64 + n1)) * F_;
#pragma unroll
    for (int t = 0; t < 4; ++t) {
      const int col = f0 + t * 16 + l16;
      v2f bx;
      bx.x = x[rowOffA + col];
      bx.y = x[rowOffB + col];
      accR[t] = wmma4(ar, bx, accR[t]);
      accI[t] = wmma4(ai, bx, accI[t]);
    }
  }
#pragma unroll
  for (int v = 0; v < 8; ++v) {
    const int k2 = m0 + half * 8 + v;
    if (k2 <= 64) {
      float s, c;                        // twiddle shared by all 4 f-tiles
      __sincosf(TWO_PI * (float)((n1 * k2) & 8191) * (1.0f / 8192.0f), &s, &c);
      const size_t rowOut = ((size_t)bn * KP + k2) * F_;
#pragma unroll
      for (int t = 0; t < 4; ++t) {
        const float r = accR[t][v], i = accI[t][v];
        float2 o;                        // multiply by (c - i s)
        o.x = r * c + i * s;
        o.y = i * c - r * s;
        buf1[rowOut + f0 + t * 16 + l16] = o;
      }
    }
  }
}

// ---------------------------------------------------------------------------
// K2: per (b, k2 in [0,64], f-block of 64):
//   T1[k1,f] = sum_n1 w64^(-k1 n1) Bm[b,n1,k2,f]          (complex matmul)
//   T1 *= Hfull[k2+128*k1, f]                              (Hermitian-extended)
//   C[b,k2,n1,f] = w8192^(+n1 k2) * sum_k1 w64^(+n1 k1) T1[k1,f]
// Wave w owns row-tile m0=(w&3)*16 and the two f-tiles of pair (w>>2).
// ---------------------------------------------------------------------------
__global__ __launch_bounds__(256) void k2_mid(const float2* __restrict__ buf1,
                                              const float* __restrict__ H,
                                              float2* __restrict__ buf2) {
  __shared__ float  Tr[64][66];
  __shared__ float  Ti[64][66];
  __shared__ float2 w64t[64];           // {cos, sin}(2*pi*p/64)
  if (threadIdx.x < 64) {
    float s, c;
    __sincosf(TWO_PI * (float)threadIdx.x * (1.0f / 64.0f), &s, &c);
    float2 e; e.x = c; e.y = s;
    w64t[threadIdx.x] = e;
  }
  __syncthreads();

  const int k2    = blockIdx.x;         // 0..64
  const int fbase = blockIdx.y * 64;
  const int b     = blockIdx.z;
  const int wave  = threadIdx.x >> 5;
  const int lane  = threadIdx.x & 31;
  const int half  = lane >> 4;
  const int l16   = lane & 15;
  const int m0    = (wave & 3) * 16;    // k1 (phase1) / n1 (phase2) tile
  const int fp    = (wave >> 2) * 32;   // f-pair base within 64-f block
  const int rowA  = m0 + l16;

  const int stepK = (rowA * 4) & 63;
  const int pa0   = (rowA * 2 * half) & 63;
  const int pb0   = (pa0 + rowA) & 63;

  // ---- phase 1: forward DFT64 over n1, filter, stash in LDS ----
  {
    int pa = pa0, pb = pb0;
    v8f accR[2] = {};
    v8f accI[2] = {};
#pragma unroll 4
    for (int kk = 0; kk < 16; ++kk) {
      const int n1a = kk * 4 + 2 * half;
      const int n1b = n1a + 1;
      const float2 ea = w64t[pa];
      const float2 eb = w64t[pb];
      pa = (pa + stepK) & 63;
      pb = (pb + stepK) & 63;
      v2f ar;  ar.x = ea.x;   ar.y = eb.x;   // Re w^-
      v2f nai; nai.x = ea.y;  nai.y = eb.y;  // +sin = -Im w^-
      v2f ai;  ai.x = -ea.y;  ai.y = -eb.y;  // Im w^-
      const size_t rowOffA = ((size_t)(b * 64 + n1a) * KP + k2) * F_;
      const size_t rowOffB = ((size_t)(b * 64 + n1b) * KP + k2) * F_;
#pragma unroll
      for (int t = 0; t < 2; ++t) {
        const int col = fbase + fp + t * 16 + l16;
        const float2 va = buf1[rowOffA + col];
        const float2 vb = buf1[rowOffB + col];
        v2f br; br.x = va.x; br.y = vb.x;
        v2f bi; bi.x = va.y; bi.y = vb.y;
        accR[t] = wmma4(ar, br, accR[t]);
        accR[t] = wmma4(nai, bi, accR[t]);
        accI[t] = wmma4(ar, bi, accI[t]);
        accI[t] = wmma4(ai, br, accI[t]);
      }
    }
#pragma unroll
    for (int v = 0; v < 8; ++v) {
      const int k1 = m0 + half * 8 + v;
      const int kf = k2 + 128 * k1;     // full-spectrum bin, 0..8128
#pragma unroll
      for (int t = 0; t < 2; ++t) {
        const int colL = fp + t * 16 + l16;
        const int col  = fbase + colL;
        float hr, hi;
        if (kf <= 4096) {
          const size_t hb = ((size_t)col * FREQ + kf) * 2;
          hr = H[hb]; hi = H[hb + 1];
        } else {                        // Hermitian extension: conj(H[8192-kf])
          const size_t hb = ((size_t)col * FREQ + (8192 - kf)) * 2;
          hr = H[hb]; hi = -H[hb + 1];
        }
        const float zr = accR[t][v], zi = accI[t][v];
        Tr[k1][colL] = zr * hr - zi * hi;
        Ti[k1][colL] = zr * hi + zi * hr;
      }
    }
  }
  __syncthreads();

  // ---- phase 2: inverse DFT64 over k1, inverse twiddle, store ----
  {
    int pa = pa0, pb = pb0;
    v8f accR[2] = {};
    v8f accI[2] = {};
#pragma unroll 4
    for (int kk = 0; kk < 16; ++kk) {
      const int k1a = kk * 4 + 2 * half;
      const int k1b = k1a + 1;
      const float2 ea = w64t[pa];
      const float2 eb = w64t[pb];
      pa = (pa + stepK) & 63;
      pb = (pb + stepK) & 63;
      v2f ar;  ar.x = ea.x;   ar.y = eb.x;   // Re w^+
      v2f ai;  ai.x = ea.y;   ai.y = eb.y;   // Im w^+
      v2f nai; nai.x = -ea.y; nai.y = -eb.y;
#pragma unroll
      for (int t = 0; t < 2; ++t) {
        const int colL = fp + t * 16 + l16;
        v2f br; br.x = Tr[k1a][colL]; br.y = Tr[k1b][colL];
        v2f bi; bi.x = Ti[k1a][colL]; bi.y = Ti[k1b][colL];
        accR[t] = wmma4(ar, br, accR[t]);
        accR[t] = wmma4(nai, bi, accR[t]);
        accI[t] = wmma4(ar, bi, accI[t]);
        accI[t] = wmma4(ai, br, accI[t]);
      }
    }
#pragma unroll
    for (int v = 0; v < 8; ++v) {
      const int n1 = m0 + half * 8 + v;
      float s, c;                        // twiddle shared by both f-tiles
      __sincosf(TWO_PI * (float)((n1 * k2) & 8191) * (1.0f / 8192.0f), &s, &c);
      const size_t rowOut = ((size_t)(b * KP + k2) * 64 + n1) * F_;
#pragma unroll
      for (int t = 0; t < 2; ++t) {
        const float r = accR[t][v], i = accI[t][v];
        float2 o;                        // multiply by (c + i s)
        o.x = r * c - i * s;
        o.y = r * s + i * c;
        buf2[rowOut + fbase + fp + t * 16 + l16] = o;
      }
    }
  }
}

// ---------------------------------------------------------------------------
// K3: per (b, n1):  y[b, n1+64*n2, f] = (1/8192) Re{ sum_k2 w128^(+n2 k2) C[b,k2,n1,f] }
// Planes k2 > 64 are conj of plane 128-k2 (only 0..64 stored).
// ---------------------------------------------------------------------------
__global__ __launch_bounds__(256) void k3_inv128(const float2* __restrict__ buf2,
                                                 float* __restrict__ y) {
  __shared__ float2 w128p[128];         // {cos, -sin}(2*pi*p/128)
  if (threadIdx.x < 128) {
    float s, c;
    __sincosf(TWO_PI * (float)threadIdx.x * (1.0f / 128.0f), &s, &c);
    float2 e; e.x = c; e.y = -s;
    w128p[threadIdx.x] = e;
  }
  __syncthreads();

  const int bn   = blockIdx.x;          // b*64 + n1
  const int b    = bn >> 6;
  const int n1   = bn & 63;
  const int f0   = blockIdx.y * 64;
  const int wave = threadIdx.x >> 5;
  const int lane = threadIdx.x & 31;
  const int half = lane >> 4;
  const int l16  = lane & 15;
  const int m0   = wave * 16;           // n2 tile
  const int rowA = m0 + l16;            // n2

  const int stepK = (rowA * 4) & 127;
  int pa = (rowA * 2 * half) & 127;
  int pb = (pa + rowA) & 127;

  v8f acc[4] = {};
#pragma unroll 4
  for (int kk = 0; kk < 32; ++kk) {
    const int k2a = kk * 4 + 2 * half;
    const int k2b = k2a + 1;
    // reflected plane + conjugation for k2 > 64 (shared by all f-tiles)
    const int ksa = (k2a <= 64) ? k2a : 128 - k2a;
    const int ksb = (k2b <= 64) ? k2b : 128 - k2b;
    const float sgA = (k2a <= 64) ? 1.0f : -1.0f;
    const float sgB = (k2b <= 64) ? 1.0f : -1.0f;
    const float2 ea = w128p[pa];
    const float2 eb = w128p[pb];
    pa = (pa + stepK) & 127;
    pb = (pb + stepK) & 127;
    v2f ar;  ar.x = ea.x; ar.y = eb.x;  // cos
    v2f nas; nas.x = ea.y; nas.y = eb.y;// -sin (pre-negated in table)
    const size_t rowOffA = ((size_t)(b * KP + ksa) * 64 + n1) * F_;
    const size_t rowOffB = ((size_t)(b * KP + ksb) * 64 + n1) * F_;
#pragma unroll
    for (int t = 0; t < 4; ++t) {
      const int col = f0 + t * 16 + l16;
      const float2 va = buf2[rowOffA + col];
      const float2 vb = buf2[rowOffB + col];
      v2f br; br.x = va.x;       br.y = vb.x;
      v2f bi; bi.x = va.y * sgA; bi.y = vb.y * sgB;
      acc[t] = wmma4(ar, br, acc[t]);   // + cos * Cr
      acc[t] = wmma4(nas, bi, acc[t]);  // - sin * Ci
    }
  }
#pragma unroll
  for (int v = 0; v < 8; ++v) {
    const int n2 = m0 + half * 8 + v;
    const size_t rowOut = (size_t)(b * T_ + (n2 * 64 + n1)) * F_;
#pragma unroll
    for (int t = 0; t < 4; ++t) {
      y[rowOut + f0 + t * 16 + l16] = acc[t][v] * (1.0f / 8192.0f);
    }
  }
}

// ---------------------------------------------------------------------------
extern "C" void kernel_launch(void* const* d_in, const int* in_sizes, int n_in,
                              void* d_out, int out_size, void* d_ws, size_t ws_size,
                              hipStream_t stream) {
  const float* x = (const float*)d_in[0];   // [B, T, F] f32
  const float* H = (const float*)d_in[1];   // [F, FREQ, 2] f32
  float* y = (float*)d_out;                 // [B, T, F] f32

  const size_t elems = (size_t)B_ * 64 * KP * F_;       // 17.04M complex each
  if (ws_size < 2 * elems * sizeof(float2)) return;      // need ~273 MB scratch
  float2* buf1 = (float2*)d_ws;                          // [B][n1][k2<=64][F]
  float2* buf2 = buf1 + elems;                           // [B][k2<=64][n1][F]

  k1_fwd128<<<dim3(B_ * 64, F_ / 64), dim3(160), 0, stream>>>(x, buf1);
  k2_mid  <<<dim3(KP, F_ / 64, B_), dim3(256), 0, stream>>>(buf1, H, buf2);
  k3_inv128<<<dim3(B_ * 64, F_ / 64), dim3(256), 0, stream>>>(buf2, y);
}